// UltraEfficientRouter_65180423685465
// MI455X (gfx1250) — compile-verified
//
#include <hip/hip_runtime.h>
#include <hip/hip_bf16.h>
#include <math.h>

typedef __attribute__((ext_vector_type(2))) float v2f;
typedef __attribute__((ext_vector_type(4))) float f4;
typedef __attribute__((ext_vector_type(8))) float v8f;

// ---------------- problem constants ----------------
#define B_  8
#define C_  256
#define R_  16
#define E_  8
#define HIN 256
#define HP  32          // pooled H,W
#define PIX (HP*HP)     // 1024 pixels per batch
#define G1_ 8
#define G2_ 4
#define EPS 1e-5f

// ---------------- workspace layout (floats) ----------------
#define XD_OFF   0u                          // pooled input   [B,C,32,32]  = 2,097,152
#define H1_OFF   2097152u                    // dwconv out -> activations (in place)
#define GN1S_OFF (H1_OFF + 2097152u)         // GN1 sum/sumsq  [B][G1][2]   = 128
#define GN1A_OFF (GN1S_OFF + 128u)           // GN1 scale/shift[B][C][2]    = 4096
#define H2_OFF   (GN1A_OFF + 4096u)          // C->R out       [B,R,1024]   = 131072
#define GN2S_OFF (H2_OFF + 131072u)          // GN2 sum/sumsq  [B][G2][2]   = 64
#define GN2A_OFF (GN2S_OFF + 64u)            // GN2 scale/shift[B][R][2]    = 512
#define POOL_OFF (GN2A_OFF + 512u)           // pooled softmax [B][E]       = 64

// fast SiLU: v_exp_f32 + v_rcp_f32, no IEEE divide chains
__device__ __forceinline__ float silu_f(float y) {
    return y * __builtin_amdgcn_rcpf(1.0f + __expf(-y));
}

// ---------------- 0: zero the atomic accumulators ----------------
__global__ void k_init(float* ws) {
    unsigned t = threadIdx.x;
    if (t < 128) ws[GN1S_OFF + t] = 0.0f;
    if (t < 64)  ws[GN2S_OFF + t] = 0.0f;
    if (t < 64)  ws[POOL_OFF + t] = 0.0f;
}

// ---------------- 1: 8x8 avg pool, streaming NT float4 loads ----------------
// one thread per pooled output element; 537 MB read -> bandwidth bound (~23us)
__global__ void k_pool(const float* __restrict__ x, float* __restrict__ ws) {
    unsigned idx = blockIdx.x * blockDim.x + threadIdx.x;   // [B*C*32*32)
    unsigned wo = idx & 31u;
    unsigned t  = idx >> 5;
    unsigned ho = t & 31u;  t >>= 5;
    unsigned c  = t & 255u;
    unsigned b  = t >> 8;
    size_t base = (((size_t)(b * C_ + c) * HIN) + (size_t)ho * 8) * HIN + (size_t)wo * 8;
    const f4* p4 = (const f4*)(x + base);
    f4 acc = (f4)(0.0f);
    #pragma unroll
    for (int r = 0; r < 8; ++r) {
        f4 a  = __builtin_nontemporal_load(p4 + (size_t)r * (HIN / 4));
        f4 b4 = __builtin_nontemporal_load(p4 + (size_t)r * (HIN / 4) + 1);
        acc += a + b4;
    }
    float s = (acc.x + acc.y + acc.z + acc.w) * (1.0f / 64.0f);
    ws[XD_OFF + idx] = s;
}

// ---------------- 2: depthwise 3x3 conv + GN1 partial stats ----------------
// block = 256 threads = 8 rows of one (b,c) 32x32 plane; 4 blocks per plane
__global__ void k_dwconv(const float* __restrict__ dw_w, float* __restrict__ ws) {
    __shared__ float s1[256];
    __shared__ float s2[256];
    unsigned bq = blockIdx.x;
    unsigned quarter = bq & 3u;
    unsigned plane   = bq >> 2;
    unsigned c = plane & 255u;
    unsigned b = plane >> 8;
    unsigned tid = threadIdx.x;
    unsigned wo = tid & 31u;
    unsigned ho = quarter * 8 + (tid >> 5);

    float wv[9];
    #pragma unroll
    for (int i = 0; i < 9; ++i) wv[i] = dw_w[c * 9 + i];

    const float* pin = ws + XD_OFF + (size_t)(b * C_ + c) * PIX;
    float acc = 0.0f;
    #pragma unroll
    for (int kh = -1; kh <= 1; ++kh) {
        int hi = (int)ho + kh;
        if (hi < 0 || hi >= HP) continue;
        #pragma unroll
        for (int kw = -1; kw <= 1; ++kw) {
            int wi = (int)wo + kw;
            if (wi < 0 || wi >= HP) continue;
            acc += wv[(kh + 1) * 3 + (kw + 1)] * pin[hi * HP + wi];
        }
    }
    ws[H1_OFF + (size_t)(b * C_ + c) * PIX + ho * HP + wo] = acc;

    // block reduction of sum / sumsq for GN1 stats
    s1[tid] = acc;
    s2[tid] = acc * acc;
    __syncthreads();
    for (unsigned off = 128; off > 0; off >>= 1) {
        if (tid < off) { s1[tid] += s1[tid + off]; s2[tid] += s2[tid + off]; }
        __syncthreads();
    }
    if (tid == 0) {
        unsigned g = c >> 5;                       // 32 channels per group
        atomicAdd(&ws[GN1S_OFF + (b * G1_ + g) * 2 + 0], s1[0]);
        atomicAdd(&ws[GN1S_OFF + (b * G1_ + g) * 2 + 1], s2[0]);
    }
}

// ---------------- 3: fold GN1 stats into per-(b,c) scale/shift ----------------
__global__ void k_gn1_affine(const float* __restrict__ gamma, const float* __restrict__ beta,
                             float* __restrict__ ws) {
    unsigned i = blockIdx.x * blockDim.x + threadIdx.x;   // [0, B*C)
    if (i >= B_ * C_) return;
    unsigned c = i & 255u, b = i >> 8;
    unsigned g = c >> 5;
    const float N = 32.0f * PIX;                           // 32768
    float sum = ws[GN1S_OFF + (b * G1_ + g) * 2 + 0];
    float sq  = ws[GN1S_OFF + (b * G1_ + g) * 2 + 1];
    float mu  = sum / N;
    float var = sq / N - mu * mu;
    float inv = rsqrtf(var + EPS);
    float sc  = gamma[c] * inv;
    ws[GN1A_OFF + i * 2 + 0] = sc;
    ws[GN1A_OFF + i * 2 + 1] = beta[c] - mu * sc;
}

// ---------------- 4: apply GN1 affine + SiLU in place (streaming f4 pass) -----
// moves all elementwise nonlinearity OUT of the WMMA loop
__global__ void k_act(float* __restrict__ ws) {
    unsigned idx = blockIdx.x * blockDim.x + threadIdx.x;  // [0, B*C*256) quads
    unsigned pq = idx & 255u;        // 256 quads of 4 pixels
    unsigned t  = idx >> 8;
    unsigned c  = t & 255u;
    unsigned b  = t >> 8;
    float sc = ws[GN1A_OFF + (b * C_ + c) * 2 + 0];
    float sh = ws[GN1A_OFF + (b * C_ + c) * 2 + 1];
    f4* p = (f4*)(ws + H1_OFF + (size_t)(b * C_ + c) * PIX) + pq;
    f4 v = *p;
    f4 o;
    o.x = silu_f(v.x * sc + sh);
    o.y = silu_f(v.y * sc + sh);
    o.z = silu_f(v.z * sc + sh);
    o.w = silu_f(v.w * sc + sh);
    *p = o;
}

// ---------------- 5: C->R 1x1 conv as WMMA f32 GEMM + GN2 partial stats -------
// GEMM: D[p, r] = sum_c act[b,c,p] * pw_w[r,c]
// M = pixels, N = R = 16, K = C = 256 (64 k-steps of 4)
// one wave handles FOUR 16-pixel tiles (64 pixels): 4 WMMAs share one B-load.
// 128 waves total; EXEC all-1s, uniform control flow (WMMA requirement).
__global__ void k_gemm_wmma(const float* __restrict__ pw_w, float* __restrict__ ws) {
    unsigned wave = (blockIdx.x * blockDim.x + threadIdx.x) >> 5;   // 0..127
    unsigned lane = threadIdx.x & 31u;
    unsigned b    = wave >> 4;            // 16 waves per batch (1024/64 px)
    unsigned p0   = (wave & 15u) << 6;    // 64-pixel base
    unsigned m    = lane & 15u;           // A row (pixel) / B col (r)
    unsigned ksel = lane >> 4;            // 0: K=kk,kk+1  1: K=kk+2,kk+3

    const float* act = ws + H1_OFF + (size_t)b * C_ * PIX;

    v8f acc0 = {}, acc1 = {}, acc2 = {}, acc3 = {};
    for (int kk = 0; kk < C_; kk += 4) {
        int c0 = kk + (int)ksel * 2;
        int c1 = c0 + 1;
        // B: pw_w^T -> B[k][n] = pw_w[n][k], n = m (same for all 4 tiles)
        v2f Bv = {pw_w[m * C_ + c0], pw_w[m * C_ + c1]};
        const float* a0p = act + (size_t)c0 * PIX + p0 + m;
        const float* a1p = act + (size_t)c1 * PIX + p0 + m;
        v2f A0 = {a0p[0],  a1p[0]};
        v2f A1 = {a0p[16], a1p[16]};
        v2f A2 = {a0p[32], a1p[32]};
        v2f A3 = {a0p[48], a1p[48]};
        acc0 = __builtin_amdgcn_wmma_f32_16x16x4_f32(false, A0, false, Bv, (short)0, acc0, false, false);
        acc1 = __builtin_amdgcn_wmma_f32_16x16x4_f32(false, A1, false, Bv, (short)0, acc1, false, false);
        acc2 = __builtin_amdgcn_wmma_f32_16x16x4_f32(false, A2, false, Bv, (short)0, acc2, false, false);
        acc3 = __builtin_amdgcn_wmma_f32_16x16x4_f32(false, A3, false, Bv, (short)0, acc3, false, false);
    }

    // D layout: VGPR v -> pixel tile_base + v (lanes 0-15) / + v + 8 (lanes 16-31)
    unsigned r = m;
    float* h2b = ws + H2_OFF + (size_t)(b * R_ + r) * PIX;
    float s = 0.0f, ss = 0.0f;
    v8f accs[4] = {acc0, acc1, acc2, acc3};
    #pragma unroll
    for (int tl = 0; tl < 4; ++tl) {
        #pragma unroll
        for (int v = 0; v < 8; ++v) {
            unsigned pix = p0 + tl * 16 + v + ksel * 8;
            float val = accs[tl][v];
            h2b[pix] = val;
            s += val; ss += val * val;
        }
    }
    unsigned g2 = r >> 2;                  // 4 channels per group
    atomicAdd(&ws[GN2S_OFF + (b * G2_ + g2) * 2 + 0], s);
    atomicAdd(&ws[GN2S_OFF + (b * G2_ + g2) * 2 + 1], ss);
}

// ---------------- 6: fold GN2 stats into per-(b,r) scale/shift ----------------
__global__ void k_gn2_affine(const float* __restrict__ gamma, const float* __restrict__ beta,
                             float* __restrict__ ws) {
    unsigned i = threadIdx.x;              // [0, B*R) = 128
    if (i >= B_ * R_) return;
    unsigned r = i & 15u, b = i >> 4;
    unsigned g = r >> 2;
    const float N = 4.0f * PIX;            // 4096
    float sum = ws[GN2S_OFF + (b * G2_ + g) * 2 + 0];
    float sq  = ws[GN2S_OFF + (b * G2_ + g) * 2 + 1];
    float mu  = sum / N;
    float var = sq / N - mu * mu;
    float inv = rsqrtf(var + EPS);
    float sc  = gamma[r] * inv;
    ws[GN2A_OFF + i * 2 + 0] = sc;
    ws[GN2A_OFF + i * 2 + 1] = beta[r] - mu * sc;
}

// ---------------- 7: R->E logits + clip + softmax + spatial mean --------------
__global__ void k_softmax_pool(const float* __restrict__ out_w, const float* __restrict__ out_b,
                               float* __restrict__ ws) {
    unsigned gid = blockIdx.x * blockDim.x + threadIdx.x;  // [0, B*1024)
    unsigned b = gid >> 10, p = gid & 1023u;
    float h[R_];
    #pragma unroll
    for (int r = 0; r < R_; ++r) {
        unsigned i = b * R_ + r;
        float v = ws[H2_OFF + (size_t)i * PIX + p];
        h[r] = silu_f(v * ws[GN2A_OFF + i * 2] + ws[GN2A_OFF + i * 2 + 1]);
    }
    float lg[E_];
    float mx = -1e30f;
    #pragma unroll
    for (int e = 0; e < E_; ++e) {
        float a = out_b[e];
        #pragma unroll
        for (int r = 0; r < R_; ++r) a += out_w[e * R_ + r] * h[r];
        a = fminf(fmaxf(a, -30.0f), 30.0f);
        lg[e] = a;
        mx = fmaxf(mx, a);
    }
    float sum = 0.0f;
    #pragma unroll
    for (int e = 0; e < E_; ++e) { float t = __expf(lg[e] - mx); lg[e] = t; sum += t; }
    float inv = __builtin_amdgcn_rcpf(sum) * (1.0f / (float)PIX);
    #pragma unroll
    for (int e = 0; e < E_; ++e) atomicAdd(&ws[POOL_OFF + b * E_ + e], lg[e] * inv);
}

// ---------------- 8: top-2 + renorm, emit outputs -----------------------------
__global__ void k_top2(const float* __restrict__ ws, float* __restrict__ vout,
                       int* __restrict__ iout) {
    unsigned b = threadIdx.x;
    if (b >= B_) return;
    float b1 = -1e30f, b2 = -1e30f;
    int i1 = 0, i2 = 0;
    for (int e = 0; e < E_; ++e) {
        float v = ws[POOL_OFF + b * E_ + e];
        if (v > b1)      { b2 = b1; i2 = i1; b1 = v; i1 = e; }
        else if (v > b2) { b2 = v;  i2 = e; }
    }
    float s = 1.0f / (b1 + b2 + 1e-6f);
    vout[b * 2 + 0] = b1 * s;
    vout[b * 2 + 1] = b2 * s;
    iout[b * 2 + 0] = i1;
    iout[b * 2 + 1] = i2;
}

// ---------------- launcher ----------------
extern "C" void kernel_launch(void* const* d_in, const int* in_sizes, int n_in,
                              void* d_out, int out_size, void* d_ws, size_t ws_size,
                              hipStream_t stream) {
    const float* x         = (const float*)d_in[0];
    const float* dw_w      = (const float*)d_in[1];
    const float* gn1_gamma = (const float*)d_in[2];
    const float* gn1_beta  = (const float*)d_in[3];
    const float* pw_w      = (const float*)d_in[4];
    const float* gn2_gamma = (const float*)d_in[5];
    const float* gn2_beta  = (const float*)d_in[6];
    const float* out_w     = (const float*)d_in[7];
    const float* out_b     = (const float*)d_in[8];
    float* ws   = (float*)d_ws;
    float* vout = (float*)d_out;            // [B,2] top-2 weights
    int*   iout = (int*)d_out + B_ * 2;     // [B,2] expert indices

    k_init<<<1, 256, 0, stream>>>(ws);
    k_pool<<<(B_ * C_ * PIX) / 256, 256, 0, stream>>>(x, ws);
    k_dwconv<<<B_ * C_ * 4, 256, 0, stream>>>(dw_w, ws);
    k_gn1_affine<<<(B_ * C_) / 256, 256, 0, stream>>>(gn1_gamma, gn1_beta, ws);
    k_act<<<(B_ * C_ * (PIX / 4)) / 256, 256, 0, stream>>>(ws);
    k_gemm_wmma<<<32, 128, 0, stream>>>(pw_w, ws);    // 128 waves, 4 tiles each
    k_gn2_affine<<<1, 128, 0, stream>>>(gn2_gamma, gn2_beta, ws);
    k_softmax_pool<<<(B_ * PIX) / 256, 256, 0, stream>>>(out_w, out_b, ws);
    k_top2<<<1, 32, 0, stream>>>(ws, vout, iout);
}